// DiscreteMMSE_33749853012536
// MI455X (gfx1250) — compile-verified
//
#include <hip/hip_runtime.h>

// Sizes fixed by the reference problem.
#define BB   32
#define NN   1024
#define DD   64
#define TT   2048
#define GG   8                 // T-chunks per batch (workgroups per batch)
#define TC   (TT / GG)         // 256 columns per workgroup
#define NB   16                // n-rows per WMMA block (= WMMA M)
#define NBP  (NB + 4)          // padded row stride (80 B: 16B-aligned, bank-spread)
#define NBLK (NN / NB)         // 64 sequential blocks
#define BN   (BB * NN)
#define BNG  (BN * GG)

typedef float v2f __attribute__((ext_vector_type(2)));
typedef float v4f __attribute__((ext_vector_type(4)));
typedef float v8f __attribute__((ext_vector_type(8)));

// ds_swizzle_b32 xor-mode (group-of-32 bitmask mode): imm = xor<<10 | or<<5 | and
#define SWIZ(v, imm) \
    __int_as_float(__builtin_amdgcn_ds_swizzle(__float_as_int(v), (imm)))

__device__ __forceinline__ float wave_max32(float v) {
    v = fmaxf(v, SWIZ(v, 0x401F));   // xor 16
    v = fmaxf(v, SWIZ(v, 0x201F));   // xor 8
    v = fmaxf(v, SWIZ(v, 0x101F));   // xor 4
    v = fmaxf(v, SWIZ(v, 0x081F));   // xor 2
    v = fmaxf(v, SWIZ(v, 0x041F));   // xor 1
    return v;
}
__device__ __forceinline__ float wave_sum32(float v) {
    v += SWIZ(v, 0x401F);
    v += SWIZ(v, 0x201F);
    v += SWIZ(v, 0x101F);
    v += SWIZ(v, 0x081F);
    v += SWIZ(v, 0x041F);
    return v;
}

// Streaming-softmax merge of (m, num, den) triples (used in combine phases).
__device__ __forceinline__ void smerge(float& m, float& num, float& den,
                                       float m2, float n2, float d2) {
    float M  = fmaxf(m, m2);
    float s1 = __expf(m - M);
    float s2 = __expf(m2 - M);
    num = num * s1 + n2 * s2;
    den = den * s1 + d2 * s2;
    m   = M;
}

// Kernel 1: one workgroup per (batch b, T-chunk g).
// Software-pipelined: WMMA for block i+1 is issued before the sequential
// softmax-scan of block i, with double-buffered LDS tiles -> 1 barrier/block.
__global__ __launch_bounds__(256)
void mmse_scan_kernel(const float* __restrict__ data,     // (B, N, D)
                      const float* __restrict__ targets,  // (B, N)
                      const float* __restrict__ W,        // (D, T)
                      float* __restrict__ wsM,            // (G, B*N)
                      float* __restrict__ wsNum,          // (G, B*N)
                      float* __restrict__ wsDen)          // (G, B*N)
{
    // xw tile stored transposed: [buf][col][row(+pad)]
    __shared__ alignas(16) float xw[2][TC][NBP];    // 40 KB
    __shared__ alignas(16) float red[2][NB][8][4];  // 4 KB, float4 per triple

    const int tid  = threadIdx.x;       // 0..255
    const int lane = tid & 31;
    const int wave = tid >> 5;          // 0..7
    const int lm   = lane & 15;         // lane % 16
    const int lh   = lane >> 4;         // lane half (0/1)

    const int b     = (int)blockIdx.x >> 3;
    const int g     = (int)blockIdx.x & (GG - 1);
    const int tbase = g * TC;           // global column base of this chunk

    const float* dataB = data + (size_t)b * NN * DD;

    // ---- Preload W B-fragments (loop-invariant) into registers: 64 VGPRs ----
    // B is 4x16 per k-step: lane holds (K=4s+2*lh+{0,1}, N=lm).
    v2f bfrag[2][16];
#pragma unroll
    for (int j = 0; j < 2; ++j) {
        const int gcol = tbase + wave * 32 + j * 16 + lm;
#pragma unroll
        for (int s = 0; s < 16; ++s) {
            const int k0 = 4 * s + 2 * lh;
            v2f bf;
            bf.x = W[(size_t)k0 * TT + gcol];
            bf.y = W[(size_t)(k0 + 1) * TT + gcol];
            bfrag[j][s] = bf;
        }
    }

    // GEMM for one 16-row block into LDS buffer `buf`.
    auto compute = [&](int blk, int buf) {
        const int n0 = blk * NB;
        // A is 16x4 per step: lane holds (M=lm, K=4s+2*lh+{0,1}).
        v2f af[16];
#pragma unroll
        for (int s = 0; s < 16; ++s) {
            const float* p = dataB + (size_t)(n0 + lm) * DD + 4 * s + 2 * lh;
            af[s] = *reinterpret_cast<const v2f*>(p);
        }
#pragma unroll
        for (int j = 0; j < 2; ++j) {
            v8f acc = {};
#pragma unroll
            for (int s = 0; s < 16; ++s) {
                acc = __builtin_amdgcn_wmma_f32_16x16x4_f32(
                    /*neg_a=*/false, af[s], /*neg_b=*/false, bfrag[j][s],
                    /*c_mod=*/(short)0, acc, /*reuse_a=*/false, /*reuse_b=*/false);
            }
            // D layout: VGPR r holds row M=r (lanes 0-15) / M=r+8 (lanes 16-31).
            const int lcol = wave * 32 + j * 16 + lm;
            v4f lo = {acc[0], acc[1], acc[2], acc[3]};
            v4f hi = {acc[4], acc[5], acc[6], acc[7]};
            *reinterpret_cast<v4f*>(&xw[buf][lcol][8 * lh])     = lo;
            *reinterpret_cast<v4f*>(&xw[buf][lcol][8 * lh + 4]) = hi;
        }
    };

    // Exclusive prefix of log-probs for this thread's column (t = tbase + tid).
    float cum = 0.0f;

    compute(0, 0);
    __syncthreads();

    for (int blk = 0; blk < NBLK; ++blk) {
        const int cur = blk & 1;
        const int n0  = blk * NB;

        // Issue next block's GEMM into the other buffer (overlaps with scan).
        if (blk + 1 < NBLK) compute(blk + 1, cur ^ 1);

        // ---- Sequential scan over the 16 rows of block `blk` ----
        v4f xr[4], tr[4];
#pragma unroll
        for (int q = 0; q < 4; ++q)
            xr[q] = *reinterpret_cast<const v4f*>(&xw[cur][tid][4 * q]);
        const float* tb = targets + b * NN + n0;
#pragma unroll
        for (int q = 0; q < 4; ++q)
            tr[q] = *reinterpret_cast<const v4f*>(tb + 4 * q);

#pragma unroll
        for (int n = 0; n < NB; ++n) {
            const float x = xr[n >> 2][n & 3];   // XW[b, n0+n, t]

            // alpha = cum (exclusive prefix).  Max-first wave reduction:
            const float m  = wave_max32(cum);
            const float e  = __expf(cum - m);    // <= 1, den >= 1: stable
            const float nu = wave_sum32(e * x);
            const float de = wave_sum32(e);
            if (lane == 0) {
                v4f t4 = {m, nu, de, 0.0f};
                *reinterpret_cast<v4f*>(&red[cur][n][wave][0]) = t4;
            }
            // Advance prefix: log_prob = -0.5 * (target - xw)^2
            // (SCALE==1 -> log_scale_sq term cancels in softmax anyway)
            const float df = tr[n >> 2][n & 3] - x;
            cum = fmaf(df, df * -0.5f, cum);
        }

        // One barrier: publishes red[cur] for combine, publishes next xw
        // buffer, and fences this block's xw reads against later overwrites.
        __syncthreads();

        // ---- Cross-wave combine, spread over all waves: 16 workers ----
        // (lane 0 and lane 16 of wave w handle n = 2w and 2w+1)
        if ((tid & 15) == 0) {
            const int n = tid >> 4;
            v4f t0 = *reinterpret_cast<const v4f*>(&red[cur][n][0][0]);
            float m = t0.x, num = t0.y, den = t0.z;
#pragma unroll
            for (int w = 1; w < 8; ++w) {
                v4f tw = *reinterpret_cast<const v4f*>(&red[cur][n][w][0]);
                smerge(m, num, den, tw.x, tw.y, tw.z);
            }
            const size_t o = (size_t)g * BN + (size_t)b * NN + n0 + n;
            wsM[o] = m; wsNum[o] = num; wsDen[o] = den;
        }
        // No second barrier needed: next block's scan writes red[cur^1].
    }
}

// Kernel 2: merge the G per-chunk partials for each (b, n) and emit preds.
__global__ __launch_bounds__(256)
void mmse_combine_kernel(const float* __restrict__ wsM,
                         const float* __restrict__ wsNum,
                         const float* __restrict__ wsDen,
                         float* __restrict__ out)
{
    const int i = (int)blockIdx.x * 256 + (int)threadIdx.x;
    if (i >= BN) return;
    float m = wsM[i], num = wsNum[i], den = wsDen[i];
#pragma unroll
    for (int g = 1; g < GG; ++g) {
        const size_t o = (size_t)g * BN + i;
        smerge(m, num, den, wsM[o], wsNum[o], wsDen[o]);
    }
    out[i] = num / den;
}

extern "C" void kernel_launch(void* const* d_in, const int* in_sizes, int n_in,
                              void* d_out, int out_size, void* d_ws, size_t ws_size,
                              hipStream_t stream) {
    (void)in_sizes; (void)n_in; (void)out_size; (void)ws_size;
    const float* data    = (const float*)d_in[0];   // (32, 1024, 64)
    const float* targets = (const float*)d_in[1];   // (32, 1024)
    const float* W       = (const float*)d_in[2];   // (64, 2048)
    float*       out     = (float*)d_out;           // (32, 1024)

    float* wsBase = (float*)d_ws;                   // needs 3 * BNG * 4 = 3 MB
    float* wsM    = wsBase;
    float* wsNum  = wsBase + (size_t)BNG;
    float* wsDen  = wsBase + (size_t)2 * BNG;

    mmse_scan_kernel<<<dim3(BB * GG), dim3(256), 0, stream>>>(
        data, targets, W, wsM, wsNum, wsDen);
    mmse_combine_kernel<<<dim3((BN + 255) / 256), dim3(256), 0, stream>>>(
        wsM, wsNum, wsDen, out);
}